// RelationUpdateLayer_89859305767507
// MI455X (gfx1250) — compile-verified
//
#include <hip/hip_runtime.h>
#include <math.h>

typedef __attribute__((ext_vector_type(2))) float v2f;
typedef __attribute__((ext_vector_type(8))) float v8f;

#define DIM 64
#define NBLK 1024

// ---------------------------------------------------------------------------
// Kernel 0: zero the workspace accumulators (colsum[64] + sumsq[64]).
// ---------------------------------------------------------------------------
__global__ void k_zero(float* __restrict__ ws) {
    if (threadIdx.x < 128) ws[threadIdx.x] = 0.f;
}

// ---------------------------------------------------------------------------
// Kernel 1: colsum[c] = sum over all N rows of e_o[:, c].
// Each thread's float4 columns are fixed (stride is a multiple of 16 float4s),
// so accumulate in registers, combine in LDS, flush with 64 atomics.
// ---------------------------------------------------------------------------
__global__ __launch_bounds__(256) void k_colsum(const float4* __restrict__ eo4,
                                                float* __restrict__ colsum,
                                                int n4) {
    __shared__ float s[DIM];
    if (threadIdx.x < DIM) s[threadIdx.x] = 0.f;
    __syncthreads();

    float a0 = 0.f, a1 = 0.f, a2 = 0.f, a3 = 0.f;
    int stride = gridDim.x * blockDim.x;
    for (int i = blockIdx.x * blockDim.x + threadIdx.x; i < n4; i += stride) {
        float4 v = eo4[i];
        a0 += v.x; a1 += v.y; a2 += v.z; a3 += v.w;
    }
    int c0 = (threadIdx.x << 2) & 63;
    atomicAdd(&s[c0 + 0], a0);
    atomicAdd(&s[c0 + 1], a1);
    atomicAdd(&s[c0 + 2], a2);
    atomicAdd(&s[c0 + 3], a3);
    __syncthreads();
    if (threadIdx.x < DIM) atomicAdd(&colsum[threadIdx.x], s[threadIdx.x]);
}

// ---------------------------------------------------------------------------
// Kernel 2: fused WMMA GEMMs + gate/relu/pair epilogue + sum-of-squares.
//
// Each wave owns a fixed 16-column tile (n = wave_id & 3) and grid-strides
// over 16-row blocks. B fragments (4 matrices x 16 ksteps x v2f = 128 VGPRs)
// are loaded ONCE per wave from global (weights are L2-resident) and stay
// register-resident — sized to fit so nothing spills.
//
// V_WMMA_F32_16X16X4_F32 layouts (ISA 7.12.2):
//   A 16x4 : lanes 0-15 rows, VGPR{0,1}=K{0,1}; lanes 16-31 = K{2,3}
//   B 4x16 : VGPR0 lanes 0-15 K=0 / lanes 16-31 K=2; VGPR1 K=1 / K=3
//   C/D    : VGPR v = rows v (lanes 0-15) and v+8 (lanes 16-31)
// ---------------------------------------------------------------------------
__global__ __launch_bounds__(256) void k_fused(
    const float* __restrict__ es, const float* __restrict__ eo,
    const float* __restrict__ V1, const float* __restrict__ V2,
    const float* __restrict__ W1, const float* __restrict__ W2,
    const float* __restrict__ w,  const float* __restrict__ colsum,
    float* __restrict__ sumsq,    float* __restrict__ out, int nTiles) {

    __shared__ float sSq[DIM];
    if (threadIdx.x < DIM) sSq[threadIdx.x] = 0.f;

    const int lane = threadIdx.x & 31;
    const int wid  = threadIdx.x >> 5;
    const int n    = wid & 3;                      // fixed 16-column tile
    const int rt0  = blockIdx.x * 2 + (wid >> 2);  // starting 16-row block
    const int rstride = gridDim.x * 2;

    const int col  = (n << 4) + (lane & 15);       // this lane's output column
    const int krow = ((lane >> 4) << 1);           // K sub-row offset for B

    // ---- Register-resident B fragments: bv[mat][k] ----
    v2f bv[4][16];
    const float* mats[4] = {V1, V2, W1, W2};
#pragma unroll
    for (int m = 0; m < 4; ++m) {
        const float* Wm = mats[m];
#pragma unroll
        for (int k = 0; k < 16; ++k) {
            const float* p = Wm + ((k << 2) + krow) * DIM + col;
            v2f b;
            b[0] = p[0];
            b[1] = p[DIM];
            bv[m][k] = b;
        }
    }

    const float pc = w[col] * colsum[col];  // pair coefficient for this column
    float sums = 0.f;                       // partial sum of squares (this col)

    const v8f vzero = {0.f, 0.f, 0.f, 0.f, 0.f, 0.f, 0.f, 0.f};

    for (int rt = rt0; rt < nTiles; rt += rstride) {
        int row0 = rt << 4;
        // A-fragment base: lane&15 selects row; lane>>4 selects K pair
        const float* ps = es + (size_t)(row0 + (lane & 15)) * DIM + krow;
        const float* po = eo + (size_t)(row0 + (lane & 15)) * DIM + krow;

        v8f accg = vzero;  // gate pre-activation  (es@V1 + eo@V2)
        v8f acct = vzero;  // tilde pre-activation (es@W1 + eo@W2)

#pragma unroll
        for (int k = 0; k < 16; ++k) {
            v2f a_s = *(const v2f*)(ps + (k << 2));
            v2f a_o = *(const v2f*)(po + (k << 2));
            accg = __builtin_amdgcn_wmma_f32_16x16x4_f32(
                false, a_s, false, bv[0][k], (short)0, accg, false, false);
            accg = __builtin_amdgcn_wmma_f32_16x16x4_f32(
                false, a_o, false, bv[1][k], (short)0, accg, false, false);
            acct = __builtin_amdgcn_wmma_f32_16x16x4_f32(
                false, a_s, false, bv[2][k], (short)0, acct, false, false);
            acct = __builtin_amdgcn_wmma_f32_16x16x4_f32(
                false, a_o, false, bv[3][k], (short)0, acct, false, false);
        }

        // Epilogue: C/D layout — VGPR v holds rows v (lanes 0-15) / v+8 (16-31)
#pragma unroll
        for (int v = 0; v < 8; ++v) {
            int row = row0 + v + ((lane >> 4) << 3);
            size_t idx = (size_t)row * DIM + col;
            float e = es[idx];
            float g = 1.f / (1.f + __expf(-accg[v]));
            float t = acct[v] + e * pc;
            t = fmaxf(t, 0.f);
            float u = e + t * g;
            out[idx] = u;
            sums += u * u;
        }
    }

    __syncthreads();  // sSq zeroed by now; all waves done accumulating
    atomicAdd(&sSq[col], sums);
    __syncthreads();
    if (threadIdx.x < DIM) atomicAdd(&sumsq[threadIdx.x], sSq[threadIdx.x]);
}

// ---------------------------------------------------------------------------
// Kernel 3: per-column L2 normalize in place: out /= max(sqrt(sumsq), eps)
// ---------------------------------------------------------------------------
__global__ __launch_bounds__(256) void k_norm(float4* __restrict__ out4,
                                              const float* __restrict__ sumsq,
                                              int n4) {
    int c0 = (threadIdx.x << 2) & 63;
    float i0 = 1.f / fmaxf(sqrtf(sumsq[c0 + 0]), 1e-12f);
    float i1 = 1.f / fmaxf(sqrtf(sumsq[c0 + 1]), 1e-12f);
    float i2 = 1.f / fmaxf(sqrtf(sumsq[c0 + 2]), 1e-12f);
    float i3 = 1.f / fmaxf(sqrtf(sumsq[c0 + 3]), 1e-12f);
    int stride = gridDim.x * blockDim.x;
    for (int i = blockIdx.x * blockDim.x + threadIdx.x; i < n4; i += stride) {
        float4 v = out4[i];
        v.x *= i0; v.y *= i1; v.z *= i2; v.w *= i3;
        out4[i] = v;
    }
}

// ---------------------------------------------------------------------------
extern "C" void kernel_launch(void* const* d_in, const int* in_sizes, int n_in,
                              void* d_out, int out_size, void* d_ws, size_t ws_size,
                              hipStream_t stream) {
    const float* e_s = (const float*)d_in[0];
    const float* e_o = (const float*)d_in[1];
    const float* V1  = (const float*)d_in[2];
    const float* V2  = (const float*)d_in[3];
    const float* W1  = (const float*)d_in[4];
    const float* W2  = (const float*)d_in[5];
    const float* w   = (const float*)d_in[6];
    float* out = (float*)d_out;

    int N = in_sizes[0] / DIM;       // 1,000,000
    int nTiles = N / 16;             // 62,500 (N divisible by 16)
    int n4 = (N * DIM) / 4;          // float4 count

    float* wsf     = (float*)d_ws;
    float* colsum  = wsf;            // [64]
    float* sumsq   = wsf + 64;       // [64]

    k_zero<<<1, 128, 0, stream>>>(wsf);
    k_colsum<<<NBLK, 256, 0, stream>>>((const float4*)e_o, colsum, n4);
    k_fused<<<NBLK, 256, 0, stream>>>(e_s, e_o, V1, V2, W1, W2, w,
                                      colsum, sumsq, out, nTiles);
    k_norm<<<NBLK, 256, 0, stream>>>((float4*)out, sumsq, n4);
}